// EdgeGCNRegressor_66013647339606
// MI455X (gfx1250) — compile-verified
//
#include <hip/hip_runtime.h>

typedef float v2f __attribute__((ext_vector_type(2)));
typedef float v8f __attribute__((ext_vector_type(8)));

#define IN_F  64
#define HID_F 128
#define OUT_F 64

// ---------------- degree / dinv ----------------

__global__ void init_deg_kernel(float* __restrict__ degP, float* __restrict__ degN, int n) {
    int i = blockIdx.x * blockDim.x + threadIdx.x;
    if (i < n) { degP[i] = 1.0f; degN[i] = 1.0f; }  // +1 self-loop weight
}

__global__ void edge_deg_kernel(const int* __restrict__ tgt, const float* __restrict__ w,
                                float* degP, float* degN, int E) {
    int e = blockIdx.x * blockDim.x + threadIdx.x;
    if (e < E) {
        float we = w[e];
        int   t  = tgt[e];
        if (we >= 0.0f) atomicAdd(&degP[t], we);
        else            atomicAdd(&degN[t], -we);
    }
}

__global__ void dinv_kernel(float* degP, float* degN, int n) {
    int i = blockIdx.x * blockDim.x + threadIdx.x;
    if (i < n) {
        float dp = degP[i]; degP[i] = (dp > 0.0f) ? rsqrtf(dp) : 0.0f;
        float dn = degN[i]; degN[i] = (dn > 0.0f) ? rsqrtf(dn) : 0.0f;
    }
}

// ---------------- acc init (fold bias) ----------------

__global__ void init_acc_kernel(float* __restrict__ accP, float* __restrict__ accN,
                                const float* __restrict__ bP, const float* __restrict__ bN,
                                int n, int C) {
    long long i = (long long)blockIdx.x * blockDim.x + threadIdx.x;
    if (i < (long long)n * C) {
        int j = (int)(i % C);
        accP[i] = bP[j];
        accN[i] = bN[j];
    }
}

// ---------------- WMMA f32 GEMM: H = X @ W^T ----------------
// X: [n,K] row-major, W: [M,K] row-major, Hout: [n,M].
// One wave computes a 16x16 tile of H via K/4 V_WMMA_F32_16X16X4_F32 ops.
// A(16x4 f32) layout: lanes 0-15 -> rows, VGPR0/1 hold K=(0,1) for half 0, K=(2,3) for half 1.
// B(4x16 f32) layout mirrors A under transpose (cols on lanes 0-15, K split across halves).
// C/D(16x16 f32): VGPR v -> row v (lanes 0-15) / row v+8 (lanes 16-31), col = lane%16.
__global__ void wmma_gemm_nt_kernel(const float* __restrict__ X,
                                    const float* __restrict__ W,
                                    float* __restrict__ Hout,
                                    int n, int K, int M) {
    int wid  = threadIdx.x >> 5;
    int lane = threadIdx.x & 31;
    int gw   = blockIdx.x * (blockDim.x >> 5) + wid;

    int colTiles = M >> 4;
    int rowTiles = (n + 15) >> 4;
    if (gw >= rowTiles * colTiles) return;          // wave-uniform exit (EXEC stays all-1)

    int rowTile = gw / colTiles;
    int colTile = gw - rowTile * colTiles;

    int half = lane >> 4;                           // 0: lanes 0-15, 1: lanes 16-31
    int lr   = lane & 15;

    int row = rowTile * 16 + lr;
    if (row >= n) row = n - 1;                      // clamp for partial tiles (store is guarded)
    int col = colTile * 16 + lr;

    const float* xrow = X + (size_t)row * K;
    const float* wrow = W + (size_t)col * K;

    v8f acc = {};
    for (int k0 = 0; k0 < K; k0 += 4) {
        v2f a = *(const v2f*)(xrow + k0 + 2 * half);   // K = k0+2h, k0+2h+1
        v2f b = *(const v2f*)(wrow + k0 + 2 * half);   // B[k][col] = W[col][k]
        acc = __builtin_amdgcn_wmma_f32_16x16x4_f32(
            /*neg_a=*/false, a, /*neg_b=*/false, b,
            /*c_mod=*/(short)0, acc, /*reuse_a=*/false, /*reuse_b=*/false);
    }

#pragma unroll
    for (int v = 0; v < 8; ++v) {
        int outRow = rowTile * 16 + v + 8 * half;
        if (outRow < n) Hout[(size_t)outRow * M + col] = acc[v];
    }
}

// ---------------- edge scatter: one wave per edge ----------------
// Exactly one of (wp, wn) is nonzero per edge -> gather only the relevant half.
template <int C>
__global__ void edge_scatter_kernel(const int* __restrict__ src, const int* __restrict__ tgt,
                                    const float* __restrict__ w,
                                    const float* __restrict__ dinvP, const float* __restrict__ dinvN,
                                    const float* __restrict__ hP, const float* __restrict__ hN,
                                    float* accP, float* accN, int E) {
    constexpr int VEC = C / 32;
    int lane = threadIdx.x & 31;
    int e = blockIdx.x * (blockDim.x >> 5) + (threadIdx.x >> 5);
    if (e >= E) return;                             // wave-uniform

    int   s  = src[e];
    int   t  = tgt[e];
    float we = w[e];

    const float* hsrc;
    float*       adst;
    float        norm;
    if (we >= 0.0f) { norm = dinvP[s] * we * dinvP[t];    hsrc = hP; adst = accP; }
    else            { norm = dinvN[s] * (-we) * dinvN[t]; hsrc = hN; adst = accN; }

    if (norm != 0.0f) {
        const float* hp = hsrc + (size_t)s * C + lane * VEC;
        float*       ap = adst + (size_t)t * C + lane * VEC;
#pragma unroll
        for (int v = 0; v < VEC; ++v)
            atomicAdd(ap + v, hp[v] * norm);
    }
}

// ---------------- combine: agg + self-loop + (already-folded bias), ReLU of (P - N) ----------------

__global__ void combine_kernel(const float* __restrict__ hP, const float* __restrict__ hN,
                               const float* __restrict__ accP, const float* __restrict__ accN,
                               const float* __restrict__ dinvP, const float* __restrict__ dinvN,
                               float* __restrict__ out, int n, int C) {
    long long i = (long long)blockIdx.x * blockDim.x + threadIdx.x;
    if (i < (long long)n * C) {
        int node = (int)(i / C);
        float dp = dinvP[node], dn = dinvN[node];
        float vp = accP[i] + hP[i] * dp * dp;
        float vn = accN[i] + hN[i] * dn * dn;
        float r  = vp - vn;
        out[i] = (r > 0.0f) ? r : 0.0f;
    }
}

// ---------------- launch ----------------

extern "C" void kernel_launch(void* const* d_in, const int* in_sizes, int n_in,
                              void* d_out, int out_size, void* d_ws, size_t ws_size,
                              hipStream_t stream) {
    const float* x   = (const float*)d_in[0];
    const int*   ei  = (const int*)d_in[1];
    const float* ew  = (const float*)d_in[2];
    const float* W1p = (const float*)d_in[3];
    const float* b1p = (const float*)d_in[4];
    const float* W1n = (const float*)d_in[5];
    const float* b1n = (const float*)d_in[6];
    const float* W2p = (const float*)d_in[7];
    const float* b2p = (const float*)d_in[8];
    const float* W2n = (const float*)d_in[9];
    const float* b2n = (const float*)d_in[10];

    int n = in_sizes[0] / IN_F;
    int E = in_sizes[2];
    const int* src = ei;
    const int* tgt = ei + E;

    float* ws = (float*)d_ws;
    size_t off = 0;
    float* degP = ws + off; off += (size_t)n;          // becomes dinvP
    float* degN = ws + off; off += (size_t)n;          // becomes dinvN
    float* hP   = ws + off; off += (size_t)n * HID_F;
    float* hN   = ws + off; off += (size_t)n * HID_F;
    float* accP = ws + off; off += (size_t)n * HID_F;
    float* accN = ws + off; off += (size_t)n * HID_F;
    float* h1   = ws + off; off += (size_t)n * HID_F;
    // layer-2 reuses layer-1 buffers (sized for HID_F >= OUT_F)
    float* h2P   = hP;
    float* h2N   = hN;
    float* acc2P = accP;
    float* acc2N = accN;

    const int T = 256;

    // ---- degrees (shared by both layers) ----
    init_deg_kernel<<<(n + T - 1) / T, T, 0, stream>>>(degP, degN, n);
    edge_deg_kernel<<<(E + T - 1) / T, T, 0, stream>>>(tgt, ew, degP, degN, E);
    dinv_kernel<<<(n + T - 1) / T, T, 0, stream>>>(degP, degN, n);

    // ---- layer 1 ----
    long long nc1 = (long long)n * HID_F;
    init_acc_kernel<<<(int)((nc1 + T - 1) / T), T, 0, stream>>>(accP, accN, b1p, b1n, n, HID_F);
    {
        int tiles = ((n + 15) / 16) * (HID_F / 16);
        int blks  = (tiles + 7) / 8;                    // 8 waves / 256-thread block
        wmma_gemm_nt_kernel<<<blks, 256, 0, stream>>>(x, W1p, hP, n, IN_F, HID_F);
        wmma_gemm_nt_kernel<<<blks, 256, 0, stream>>>(x, W1n, hN, n, IN_F, HID_F);
    }
    edge_scatter_kernel<HID_F><<<(E + 7) / 8, 256, 0, stream>>>(src, tgt, ew, degP, degN,
                                                                hP, hN, accP, accN, E);
    combine_kernel<<<(int)((nc1 + T - 1) / T), T, 0, stream>>>(hP, hN, accP, accN,
                                                               degP, degN, h1, n, HID_F);

    // ---- layer 2 ----
    long long nc2 = (long long)n * OUT_F;
    init_acc_kernel<<<(int)((nc2 + T - 1) / T), T, 0, stream>>>(acc2P, acc2N, b2p, b2n, n, OUT_F);
    {
        int tiles = ((n + 15) / 16) * (OUT_F / 16);
        int blks  = (tiles + 7) / 8;
        wmma_gemm_nt_kernel<<<blks, 256, 0, stream>>>(h1, W2p, h2P, n, HID_F, OUT_F);
        wmma_gemm_nt_kernel<<<blks, 256, 0, stream>>>(h1, W2n, h2N, n, HID_F, OUT_F);
    }
    edge_scatter_kernel<OUT_F><<<(E + 7) / 8, 256, 0, stream>>>(src, tgt, ew, degP, degN,
                                                                h2P, h2N, acc2P, acc2N, E);
    combine_kernel<<<(int)((nc2 + T - 1) / T), T, 0, stream>>>(h2P, h2N, acc2P, acc2N,
                                                               degP, degN, (float*)d_out, n, OUT_F);
}